// DetClsBenchEval_42477226557705
// MI455X (gfx1250) — compile-verified
//
#include <hip/hip_runtime.h>
#include <hip/hip_bf16.h>
#include <stdint.h>

// ---------------------------------------------------------------------------
// EfficientDet post-processing for MI455X (gfx1250, wave32).
// Two full streaming passes over the 283MB of class logits:
//   pass 1: per-image 13-bit radix histogram (LDS-private, merged once)
//   pass 2: compaction against the exact 13-bit threshold (+tiny tie bin)
// then exact tie resolution, decode+sigmoid on 16x5000, and a fully
// LDS-resident 100-round NMS per image.
// CDNA5 paths: global_load_async_to_lds_b128 + s_wait_asynccnt staging,
// global_prefetch_b8 in the streaming loops, 128KB LDS NMS state.
// ---------------------------------------------------------------------------

namespace {
constexpr int BATCH = 16, NCLS = 90, TOPK = 5000, NDET = 100, EQCAP = 4096;
constexpr int NBIN = 8192;  // 13-bit histogram

// workspace layout, in u32 units
constexpr size_t OFF_H13  = 0;                                   // [16][8192]
constexpr size_t OFF_META = OFF_H13  + (size_t)BATCH * NBIN;     // [16][16]
constexpr size_t OFF_SELS = OFF_META + (size_t)BATCH * 16;       // [16][5000] f32
constexpr size_t OFF_SELI = OFF_SELS + (size_t)BATCH * TOPK;     // [16][5000] u32
constexpr size_t OFF_EQK  = OFF_SELI + (size_t)BATCH * TOPK;     // [16][4096] u32
constexpr size_t OFF_EQI  = OFF_EQK  + (size_t)BATCH * EQCAP;    // [16][4096] u32
constexpr size_t OFF_NMS  = OFF_EQI  + (size_t)BATCH * EQCAP;    // [16][55008] f32
constexpr size_t NMS_STRIDE = 55008;  // nb0..3,area,score (30000) box0..3,cls (25000) +pad
constexpr size_t ZERO_BYTES = OFF_SELS * 4;  // histogram + meta/counters zeroed per launch
// meta fields: 0=t13 1=cntAbove 2=needed 3=ctrG 4=ctrE
}

// order-preserving float->u32 key (bigger float => bigger key)
static __device__ __forceinline__ unsigned fkey(float f) {
  unsigned u = __float_as_uint(f);
  return (u & 0x80000000u) ? ~u : (u | 0x80000000u);
}
static __device__ __forceinline__ float keyToF(unsigned k) {
  unsigned u = (k & 0x80000000u) ? (k ^ 0x80000000u) : ~k;
  return __uint_as_float(u);
}

// ---- pass 1: per-image 13-bit histogram (float4 streaming, LDS-private) ----
__global__ __launch_bounds__(256) void k_hist13(const float* __restrict__ src,
                                                int n4, unsigned* __restrict__ h13) {
  const int b = blockIdx.y;
  __shared__ unsigned lh[NBIN];
  for (int i = threadIdx.x; i < NBIN; i += blockDim.x) lh[i] = 0u;
  __syncthreads();
  const float4* p4 = (const float4*)(src + (size_t)b * (size_t)n4 * 4);
  const int stride = blockDim.x * gridDim.x;
  for (int i = blockIdx.x * blockDim.x + threadIdx.x; i < n4; i += stride) {
    __builtin_prefetch(p4 + i + 4 * stride, 0, 0);   // global_prefetch_b8
    const float4 v = p4[i];
    atomicAdd(&lh[fkey(v.x) >> 19], 1u);
    atomicAdd(&lh[fkey(v.y) >> 19], 1u);
    atomicAdd(&lh[fkey(v.z) >> 19], 1u);
    atomicAdd(&lh[fkey(v.w) >> 19], 1u);
  }
  __syncthreads();
  for (int i = threadIdx.x; i < NBIN; i += blockDim.x)
    if (lh[i]) atomicAdd(&h13[(size_t)b * NBIN + i], lh[i]);
}

// ---- find 13-bit threshold bin per image -----------------------------------
__global__ __launch_bounds__(256) void k_scan13(const unsigned* __restrict__ h13,
                                                unsigned* __restrict__ meta) {
  const int b = blockIdx.x, t = threadIdx.x;
  __shared__ unsigned csum[256], csuf[256];
  const unsigned* h = h13 + (size_t)b * NBIN;
  unsigned s = 0;
  for (int i = 0; i < 32; ++i) s += h[t * 32 + i];
  csum[t] = s;
  __syncthreads();
  if (t == 0) {
    unsigned run = 0;
    for (int c = 255; c >= 0; --c) { csuf[c] = run; run += csum[c]; }
  }
  __syncthreads();
  unsigned cum = csuf[t];
  for (int bin = t * 32 + 31; bin >= t * 32; --bin) {
    const unsigned c = h[bin];
    if (cum < (unsigned)TOPK && cum + c >= (unsigned)TOPK) {
      meta[b * 16 + 0] = (unsigned)bin;          // 13-bit threshold prefix
      meta[b * 16 + 1] = cum;                    // strictly-above count
      meta[b * 16 + 2] = (unsigned)TOPK - cum;   // needed from tie bin
    }
    cum += c;
  }
}

// ---- pass 2: compaction (float4 streaming) ---------------------------------
__global__ __launch_bounds__(256) void k_compact(const float* __restrict__ src,
                                                 int n4, int hw2, int levelOff,
                                                 unsigned* __restrict__ meta,
                                                 float* __restrict__ selS,
                                                 unsigned* __restrict__ selI,
                                                 unsigned* __restrict__ eqK,
                                                 unsigned* __restrict__ eqI) {
  const int b = blockIdx.y;
  const unsigned t13 = meta[b * 16 + 0];
  const float4* p4 = (const float4*)(src + (size_t)b * (size_t)n4 * 4);
  const int stride = blockDim.x * gridDim.x;
  for (int i = blockIdx.x * blockDim.x + threadIdx.x; i < n4; i += stride) {
    __builtin_prefetch(p4 + i + 4 * stride, 0, 0);
    const float4 v = p4[i];
    const float vv[4] = {v.x, v.y, v.z, v.w};
#pragma unroll
    for (int c = 0; c < 4; ++c) {
      const unsigned key = fkey(vv[c]);
      const unsigned k13 = key >> 19;
      if (k13 < t13) continue;
      // tensor index e = ch*hw*hw + y*hw + x ; ch = a*90 + cls
      const int e = i * 4 + c;
      const int ch = e / hw2, pix = e % hw2;
      const int a = ch / NCLS, k = ch % NCLS;
      const unsigned anchor = (unsigned)levelOff + (unsigned)pix * 9u + (unsigned)a;
      const unsigned packed = anchor * (unsigned)NCLS + (unsigned)k;
      if (k13 > t13) {
        const unsigned slot = atomicAdd(&meta[b * 16 + 3], 1u);
        selS[(size_t)b * TOPK + slot] = vv[c];
        selI[(size_t)b * TOPK + slot] = packed;
      } else {
        const unsigned e2 = atomicAdd(&meta[b * 16 + 4], 1u);
        if (e2 < (unsigned)EQCAP) {
          eqK[(size_t)b * EQCAP + e2] = key;
          eqI[(size_t)b * EQCAP + e2] = packed;
        }
      }
    }
  }
}

// ---- exact tie resolution by rank (async-staged into LDS) ------------------
__global__ __launch_bounds__(256) void k_finalize(unsigned* __restrict__ meta,
                                                  const unsigned* __restrict__ eqK,
                                                  const unsigned* __restrict__ eqI,
                                                  float* __restrict__ selS,
                                                  unsigned* __restrict__ selI) {
  const int b = blockIdx.x, t = threadIdx.x;
  const int nEq = min((int)meta[b * 16 + 4], EQCAP);
  const int needed = (int)meta[b * 16 + 2];
  const int base = (int)meta[b * 16 + 1];
  __shared__ __align__(16) unsigned sk[EQCAP];
  __shared__ __align__(16) unsigned si[EQCAP];

  // stage both 16KB candidate arrays with async global->LDS b128 copies
  const unsigned* gk = eqK + (size_t)b * EQCAP;
  const unsigned* gi = eqI + (size_t)b * EQCAP;
  for (int c = t; c < EQCAP / 4; c += blockDim.x) {
    unsigned la0 = (unsigned)(unsigned long long)(sk + (size_t)c * 4);
    unsigned long long ga0 = (unsigned long long)(gk + (size_t)c * 4);
    asm volatile("global_load_async_to_lds_b128 %0, %1, off"
                 :: "v"(la0), "v"(ga0) : "memory");
    unsigned la1 = (unsigned)(unsigned long long)(si + (size_t)c * 4);
    unsigned long long ga1 = (unsigned long long)(gi + (size_t)c * 4);
    asm volatile("global_load_async_to_lds_b128 %0, %1, off"
                 :: "v"(la1), "v"(ga1) : "memory");
  }
  asm volatile("s_wait_asynccnt 0x0" ::: "memory");

  for (int r = t; r < needed; r += blockDim.x) {  // robust default fill
    selS[(size_t)b * TOPK + base + r] = -1e30f;
    selI[(size_t)b * TOPK + base + r] = 0u;
  }
  __syncthreads();

  for (int i = t; i < nEq; i += blockDim.x) {
    const unsigned ki = sk[i];
    int rank = 0;
    for (int j = 0; j < nEq; ++j) {
      const unsigned kj = sk[j];
      rank += (kj > ki) || (kj == ki && j < i);
    }
    if (rank < needed) {
      selS[(size_t)b * TOPK + base + rank] = keyToF(ki);
      selI[(size_t)b * TOPK + base + rank] = si[i];
    }
  }
}

// ---- decode + sigmoid on the 16x5000 survivors -----------------------------
__global__ __launch_bounds__(256) void k_decode(const float* __restrict__ selS,
                                                const unsigned* __restrict__ selI,
                                                const float* __restrict__ anchors,
                                                const float* __restrict__ bx0,
                                                const float* __restrict__ bx1,
                                                const float* __restrict__ bx2,
                                                const float* __restrict__ bx3,
                                                const float* __restrict__ bx4,
                                                float* __restrict__ nmsA) {
  const int g = blockIdx.x * blockDim.x + threadIdx.x;
  if (g >= BATCH * TOPK) return;
  const int b = g / TOPK, s = g % TOPK;
  const unsigned packed = selI[g];
  const float logit = selS[g];
  const unsigned p = packed / NCLS, k = packed % NCLS;

  int li, off, hw;
  if      (p < 36864u) { li = 0; off = 0;     hw = 64; }
  else if (p < 46080u) { li = 1; off = 36864; hw = 32; }
  else if (p < 48384u) { li = 2; off = 46080; hw = 16; }
  else if (p < 48960u) { li = 3; off = 48384; hw = 8;  }
  else                 { li = 4; off = 48960; hw = 4;  }
  const int q = (int)p - off, cell = q / 9, a = q % 9;
  const float* bo = (li == 0) ? bx0 : (li == 1) ? bx1 : (li == 2) ? bx2 : (li == 3) ? bx3 : bx4;
  const size_t hw2 = (size_t)hw * hw;
  const size_t ibase = (size_t)b * 36 * hw2 + (size_t)cell;  // y*hw+x == cell (row-major)
  const float ty = bo[ibase + (size_t)(a * 4 + 0) * hw2];
  const float tx = bo[ibase + (size_t)(a * 4 + 1) * hw2];
  const float th = bo[ibase + (size_t)(a * 4 + 2) * hw2];
  const float tw = bo[ibase + (size_t)(a * 4 + 3) * hw2];

  const float4 an = ((const float4*)anchors)[p];  // y1,x1,y2,x2
  const float ya = (an.x + an.z) * 0.5f, xa = (an.y + an.w) * 0.5f;
  const float ha = an.z - an.x, wa = an.w - an.y;
  const float hh = expf(th) * ha, ww = expf(tw) * wa;
  const float yc = ty * ha + ya, xc = tx * wa + xa;
  const float b0 = yc - hh * 0.5f, b1 = xc - ww * 0.5f;
  const float b2 = yc + hh * 0.5f, b3 = xc + ww * 0.5f;
  const float coff = (float)k * 10000.0f;  // class-offset NMS trick

  float* img = nmsA + (size_t)b * NMS_STRIDE;
  img[          s] = b0 + coff;
  img[ 5000u + s] = b1 + coff;
  img[10000u + s] = b2 + coff;
  img[15000u + s] = b3 + coff;
  img[20000u + s] = (b2 - b0) * (b3 - b1);           // area (offset cancels)
  img[25000u + s] = 1.0f / (1.0f + expf(-logit));    // sigmoid
  img[30000u + s] = b0;
  img[35000u + s] = b1;
  img[40000u + s] = b2;
  img[45000u + s] = b3;
  img[50000u + s] = (float)k;
}

// ---- per-image NMS, fully LDS-resident (CDNA5 async global->LDS staging) ---
__global__ __launch_bounds__(1024) void k_nms(const float* __restrict__ nmsA,
                                              const float* __restrict__ scales,
                                              float* __restrict__ out) {
  extern __shared__ float sm[];
  float* snb0 = sm;
  float* snb1 = sm + 5000;
  float* snb2 = sm + 10000;
  float* snb3 = sm + 15000;
  float* sarea = sm + 20000;
  float* sscore = sm + 25000;
  float* redV = sm + 30000;
  int*   redI = (int*)(sm + 31024);

  const int b = blockIdx.x, tid = threadIdx.x;
  const float* img = nmsA + (size_t)b * NMS_STRIDE;

  // Async copy 30000 f32 (nb/area/score) global->LDS: 7500 x b128 chunks.
  for (int c = tid; c < 7500; c += blockDim.x) {
    unsigned lds_addr = (unsigned)(unsigned long long)(sm + (size_t)c * 4);
    unsigned long long gaddr = (unsigned long long)(img + (size_t)c * 4);
    asm volatile("global_load_async_to_lds_b128 %0, %1, off"
                 :: "v"(lds_addr), "v"(gaddr) : "memory");
  }
  asm volatile("s_wait_asynccnt 0x0" ::: "memory");
  __syncthreads();

  const float scale = scales[b];

  for (int it = 0; it < NDET; ++it) {
    // argmax over 5000 scores (first-index tie-break, matches jnp.argmax)
    float bv = -1e30f; int bi = 0;
    for (int s = tid; s < TOPK; s += 1024) {
      const float v = sscore[s];
      if (v > bv) { bv = v; bi = s; }
    }
    redV[tid] = bv; redI[tid] = bi;
    __syncthreads();
    for (int off = 512; off > 0; off >>= 1) {
      if (tid < off) {
        const float v2 = redV[tid + off]; const int i2 = redI[tid + off];
        if (v2 > redV[tid] || (v2 == redV[tid] && i2 < redI[tid])) {
          redV[tid] = v2; redI[tid] = i2;
        }
      }
      __syncthreads();
    }
    const int imax = redI[0];
    const float vmax = redV[0];
    const float c0 = snb0[imax], c1 = snb1[imax], c2 = snb2[imax], c3 = snb3[imax];
    const float ca = sarea[imax];

    if (tid == 0) {
      float* o = out + ((size_t)b * NDET + it) * 6;
      if (vmax > -0.5f) {
        o[0] = img[30000u + imax] * scale;
        o[1] = img[35000u + imax] * scale;
        o[2] = img[40000u + imax] * scale;
        o[3] = img[45000u + imax] * scale;
        o[4] = vmax;
        o[5] = img[50000u + imax] + 1.0f;
      } else {
        o[0] = o[1] = o[2] = o[3] = o[4] = o[5] = 0.0f;
      }
    }

    for (int s = tid; s < TOPK; s += 1024) {
      const float yy1 = fmaxf(snb0[s], c0), xx1 = fmaxf(snb1[s], c1);
      const float yy2 = fminf(snb2[s], c2), xx2 = fminf(snb3[s], c3);
      const float inter = fmaxf(yy2 - yy1, 0.0f) * fmaxf(xx2 - xx1, 0.0f);
      const float iou = inter / (sarea[s] + ca - inter + 1e-8f);
      if (iou > 0.5f || s == imax) sscore[s] = -1.0f;
    }
    __syncthreads();
  }
}

// ---------------------------------------------------------------------------
extern "C" void kernel_launch(void* const* d_in, const int* in_sizes, int n_in,
                              void* d_out, int out_size, void* d_ws, size_t ws_size,
                              hipStream_t stream) {
  // setup_inputs() interleaves cls/box per level:
  const float* cls[5]; const float* box[5];
  for (int li = 0; li < 5; ++li) {
    cls[li] = (const float*)d_in[2 * li + 0];
    box[li] = (const float*)d_in[2 * li + 1];
  }
  const float* anchors = (const float*)d_in[10];
  const float* scales  = (const float*)d_in[11];
  float* out = (float*)d_out;

  unsigned* W = (unsigned*)d_ws;
  unsigned* h13  = W + OFF_H13;
  unsigned* meta = W + OFF_META;
  float*    selS = (float*)(W + OFF_SELS);
  unsigned* selI = W + OFF_SELI;
  unsigned* eqK  = W + OFF_EQK;
  unsigned* eqI  = W + OFF_EQI;
  float*    nmsA = (float*)(W + OFF_NMS);

  static const int HW[5]   = {64, 32, 16, 8, 4};
  static const int AOFF[5] = {0, 36864, 46080, 48384, 48960};

  hipMemsetAsync(d_ws, 0, ZERO_BYTES, stream);  // histogram + meta/counters

  // pass 1: big blocks so the 8192-bin merge stays ~12% of element traffic
  for (int li = 0; li < 5; ++li) {
    const int n4 = (810 * HW[li] * HW[li]) >> 2;
    int bpi = (n4 + 16383) / 16384; if (bpi > 64) bpi = 64;
    k_hist13<<<dim3(bpi, BATCH), 256, 0, stream>>>(cls[li], n4, h13);
  }
  k_scan13<<<BATCH, 256, 0, stream>>>(h13, meta);

  // pass 2: compaction (no merge -> more blocks for occupancy)
  for (int li = 0; li < 5; ++li) {
    const int n4 = (810 * HW[li] * HW[li]) >> 2;
    int bpi = (n4 + 4095) / 4096; if (bpi > 256) bpi = 256;
    k_compact<<<dim3(bpi, BATCH), 256, 0, stream>>>(cls[li], n4, HW[li] * HW[li],
                                                    AOFF[li], meta, selS, selI, eqK, eqI);
  }
  k_finalize<<<BATCH, 256, 0, stream>>>(meta, eqK, eqI, selS, selI);

  // decode + sigmoid
  const int ndec = BATCH * TOPK;
  k_decode<<<(ndec + 255) / 256, 256, 0, stream>>>(selS, selI, anchors,
                                                   box[0], box[1], box[2], box[3], box[4],
                                                   nmsA);

  // NMS: 120KB candidate state + 8KB reduction scratch in LDS
  const size_t nms_lds = (30000 + 1024 + 1024) * sizeof(float);
  k_nms<<<BATCH, 1024, nms_lds, stream>>>(nmsA, scales, out);
}